// ViterbiLoss_4896262718142
// MI455X (gfx1250) — compile-verified
//
#include <hip/hip_runtime.h>
#include <stdint.h>

#define BB 128
#define TT 512
#define KK 20
#define KK2 400          // K*K floats = 1600 bytes per (b,t) tile
#define START_TAG 18
#define END_TAG 19
#define DEPTH 8          // LDS ring depth (tiles in flight per wave)

// ---------------- CDNA5 async global->LDS helpers ----------------
#if defined(__AMDGCN__)
  #if __has_builtin(__builtin_amdgcn_global_load_async_to_lds_b128)
    #define ASYNC_USE_BUILTIN 1
  #else
    #define ASYNC_USE_BUILTIN 0
    #warning "builtin global_load_async_to_lds_b128 not found: using inline asm fallback"
  #endif
  #if __has_builtin(__builtin_amdgcn_s_wait_asynccnt)
    #define WAIT_ASYNC(n) __builtin_amdgcn_s_wait_asynccnt(n)
  #else
    #define WAIT_ASYNC(n) asm volatile("s_wait_asynccnt %0" :: "i"(n) : "memory")
  #endif
#else
  #define ASYNC_USE_BUILTIN 0
  #define WAIT_ASYNC(n) ((void)0)
#endif

typedef int v4i __attribute__((vector_size(4 * sizeof(int))));
typedef __attribute__((address_space(1))) v4i* gv4_ptr;   // global
typedef __attribute__((address_space(3))) v4i* lv4_ptr;   // LDS

__device__ __forceinline__ void async_cp16(const void* g, void* l) {
#if defined(__AMDGCN__)
#if ASYNC_USE_BUILTIN
  __builtin_amdgcn_global_load_async_to_lds_b128(
      (gv4_ptr)g, (lv4_ptr)l, /*imm offset*/0, /*cpol*/0);
#else
  unsigned lds_off =
      (unsigned)(unsigned long long)(__attribute__((address_space(3))) char*)l;
  asm volatile("global_load_async_to_lds_b128 %0, %1, off"
               :: "v"(lds_off), "v"(g) : "memory");
#endif
#else
  (void)g; (void)l;
#endif
}

// Copy one 1600-byte tile with exactly 4 async instructions (constant ASYNCcnt
// accounting: +4 per tile). Lanes 0..31 move 3*512B, lanes 0..3 move last 64B.
__device__ __forceinline__ void issue_tile(const float* g, float* l, int lane) {
  const char* gp = (const char*)g + lane * 16;
  char*       lp = (char*)l       + lane * 16;
  async_cp16(gp,        lp);
  async_cp16(gp + 512,  lp + 512);
  async_cp16(gp + 1024, lp + 1024);
  if (lane < 4) async_cp16(gp + 1536, lp + 1536);
}

// ---------------- CRF forward scan: one wave32 per batch element ----------------
__global__ void __launch_bounds__(32)
crf_forward_kernel(const float* __restrict__ scores,
                   const int*   __restrict__ lengths,
                   float*       __restrict__ ws /* [BB] final[b,END_TAG] */) {
  // ring of DEPTH tiles; +16 floats pad so lanes 20..31 reading i*20+lane stay in-bounds
  __shared__ float tile[DEPTH * KK2 + 16];

  const int b    = blockIdx.x;
  const int lane = threadIdx.x;
  const float* base = scores + (size_t)b * TT * KK2;
  const int len    = lengths[b];       // >= 1 by construction
  const int nsteps = len - 1;          // steps t = 1 .. len-1

  float carry = -__builtin_inff();
  if (lane < KK) carry = base[START_TAG * KK + lane];   // init = scores[b,0,START,:]

  if (nsteps > 0) {
    // prologue: fill the ring (clamped so the count is always DEPTH tiles = DEPTH*4 ops)
    for (int p = 0; p < DEPTH; ++p) {
      int t = 1 + p; if (t > nsteps) t = nsteps;
      issue_tile(base + (size_t)t * KK2, &tile[p * KK2], lane);
    }

    for (int s = 1; s <= nsteps; ++s) {
      const int slot = (s - 1) & (DEPTH - 1);

      // oldest tile (for step s) is complete when <= (DEPTH-1)*4 groups remain in flight
      WAIT_ASYNC((DEPTH - 1) * 4);
      asm volatile("" ::: "memory");   // keep LDS reads below the wait

      // shared stabilizer m = max_i carry[i] (lanes >=20 hold -inf)
      float m = carry;
      #pragma unroll
      for (int off = 16; off > 0; off >>= 1)
        m = fmaxf(m, __shfl_xor(m, off, 32));

      // lane j: new_j = m + log( sum_i exp(s[i][j] + carry[i] - m) )
      float acc = 0.0f;
      const float* tp = &tile[slot * KK2];
      #pragma unroll
      for (int i = 0; i < KK; ++i) {
        float ci  = __shfl(carry, i, 32);       // broadcast carry[i]
        float sij = tp[i * KK + lane];          // conflict-free: consecutive dwords per row
        acc += __expf(sij + (ci - m));
      }
      float nc = m + __logf(acc);
      if (lane < KK) carry = nc;

      // refill the just-consumed slot with tile s+DEPTH (address-clamped dummy at the tail)
      int tn = s + DEPTH; if (tn > nsteps) tn = nsteps;
      issue_tile(base + (size_t)tn * KK2, &tile[slot * KK2], lane);
    }
    WAIT_ASYNC(0);
  }

  if (lane == END_TAG) ws[b] = carry;
}

// ---------------- gold score: one wave32 per batch element ----------------
__global__ void __launch_bounds__(32)
gold_kernel(const float* __restrict__ scores,
            const int*   __restrict__ targets,
            const int*   __restrict__ lengths,
            float*       __restrict__ ws /* [BB] gold partial per b */) {
  const int b    = blockIdx.x;
  const int lane = threadIdx.x;
  const int len  = lengths[b];

  float sum = 0.0f;
  for (int t = lane; t < len; t += 32) {
    int tgt  = targets[b * TT + t];
    int prev = (t == 0) ? 0 : targets[b * TT + t - 1];
    sum += scores[((size_t)b * TT + t) * KK2 + prev * KK + tgt];
  }
  #pragma unroll
  for (int off = 16; off > 0; off >>= 1)
    sum += __shfl_xor(sum, off, 32);
  if (lane == 0) ws[b] = sum;
}

// ---------------- finalize: deterministic fixed-order reduction ----------------
__global__ void __launch_bounds__(32)
finalize_kernel(const float* __restrict__ ws, float* __restrict__ out) {
  const int lane = threadIdx.x;
  float a = 0.0f, g = 0.0f;
  for (int i = lane; i < BB; i += 32) { a += ws[i]; g += ws[BB + i]; }
  float v = a - g;
  #pragma unroll
  for (int off = 16; off > 0; off >>= 1)
    v += __shfl_xor(v, off, 32);
  if (lane == 0) out[0] = v / (float)BB;
}

extern "C" void kernel_launch(void* const* d_in, const int* in_sizes, int n_in,
                              void* d_out, int out_size, void* d_ws, size_t ws_size,
                              hipStream_t stream) {
  const float* scores  = (const float*)d_in[0];   // [B,T,K,K] f32
  const int*   targets = (const int*)  d_in[1];   // [B,T] i32
  const int*   lengths = (const int*)  d_in[2];   // [B] i32
  float* out = (float*)d_out;                     // scalar f32
  float* ws  = (float*)d_ws;                      // ws[0..127]=final, ws[128..255]=gold

  crf_forward_kernel<<<BB, 32, 0, stream>>>(scores, lengths, ws);
  gold_kernel       <<<BB, 32, 0, stream>>>(scores, targets, lengths, ws + BB);
  finalize_kernel   <<<1,  32, 0, stream>>>(ws, out);
}